// Classifier_7421703487683
// MI455X (gfx1250) — compile-verified
//
#include <hip/hip_runtime.h>
#include <hip/hip_bf16.h>
#include <math.h>

#define N_NODES 100000
#define N_EDGES 800000
#define HID 64
#define EMB 128
#define HOPS 3
#define N_GRAPHS 64

typedef __attribute__((ext_vector_type(2))) float v2f;
typedef __attribute__((ext_vector_type(8))) float v8f;

// ---------------- elementwise / setup kernels ----------------

__global__ __launch_bounds__(256) void k_zero(float* p, int n) {
    int i = blockIdx.x * 256 + threadIdx.x;
    if (i < n) p[i] = 0.f;
}

__global__ __launch_bounds__(256) void k_degree(const int* __restrict__ dst, float* __restrict__ deg) {
    int e = blockIdx.x * 256 + threadIdx.x;
    if (e < N_EDGES) atomicAdd(&deg[dst[e]], 1.0f);
}

__global__ __launch_bounds__(256) void k_degnorm(float* deg) {
    int n = blockIdx.x * 256 + threadIdx.x;
    if (n < N_NODES) deg[n] = rsqrtf(fmaxf(deg[n], 1.0f));
}

// Start of a TAGConv layer: feats[:,0:64] = x, feats[:,64:256] = 0, xs = x * deg_norm
__global__ __launch_bounds__(256) void k_prep(const float* __restrict__ x,
                                              const float* __restrict__ dn,
                                              float* __restrict__ feats,
                                              float* __restrict__ xs) {
    int i = blockIdx.x * 256 + threadIdx.x;       // over N_NODES*256
    int n = i >> 8, f = i & 255;
    if (f < 64) {
        float v = x[(n << 6) + f];
        feats[i] = v;
        xs[(n << 6) + f] = v * dn[n];
    } else {
        feats[i] = 0.f;
    }
}

// One hop of message passing: feats[dst, hopoff + c] += xs[src, c]   (float4 per thread)
__global__ __launch_bounds__(256) void k_scatter(const int* __restrict__ src,
                                                 const int* __restrict__ dst,
                                                 const float* __restrict__ xs,
                                                 float* __restrict__ feats,
                                                 int hopoff) {
    int i = blockIdx.x * 256 + threadIdx.x;       // over N_EDGES*16
    if (i >= N_EDGES * 16) return;
    int e = i >> 4;
    int c = (i & 15) << 2;
    int s = src[e], d = dst[e];
    const float4 v = *(const float4*)(xs + (s << 6) + c);
    float* o = feats + d * 256 + hopoff + c;
    atomicAdd(o + 0, v.x);
    atomicAdd(o + 1, v.y);
    atomicAdd(o + 2, v.z);
    atomicAdd(o + 3, v.w);
}

// After scatter: x_k = deg_norm * sum  (store into feats), xs = x_k * deg_norm for next hop
__global__ __launch_bounds__(256) void k_posthop(float* __restrict__ feats,
                                                 const float* __restrict__ dn,
                                                 float* __restrict__ xs,
                                                 int hopoff) {
    int i = blockIdx.x * 256 + threadIdx.x;       // over N_NODES*64
    int n = i >> 6, f = i & 63;
    int idx = n * 256 + hopoff + f;
    float s = feats[idx];
    float xk = s * dn[n];
    feats[idx] = xk;
    xs[i] = xk * dn[n];
}

// ---------------- WMMA f32 GEMM: C[M,N] = relu?(A[M,K] @ B[K,N] + bias) ----------------
// One wave per 16x16 tile, V_WMMA_F32_16X16X4_F32 with K-step 4.
// A-layout (16x4 f32): lanes 0-15 row M=l, VGPR0=K0 VGPR1=K1; lanes 16-31: K2,K3.
// B-layout (4x16 f32): VGPR0: K=0 (lanes 0-15) / K=2 (lanes 16-31); VGPR1: K=1 / K=3.
// C-layout: VGPR r: M=r (lanes 0-15), M=r+8 (lanes 16-31), N=lane%16.
__global__ __launch_bounds__(32) void k_wmma_gemm(const float* __restrict__ A, int lda,
                                                  const float* __restrict__ B, int ldb,
                                                  const float* __restrict__ bias,
                                                  float* __restrict__ C, int ldc,
                                                  int K, int relu) {
    int lane = threadIdx.x;
    int half = lane >> 4;          // 0 or 1
    int l16  = lane & 15;
    int row  = blockIdx.x * 16 + l16;
    int col  = blockIdx.y * 16 + l16;
    int koff = half << 1;          // 0 or 2

    v8f acc = {};
    const float* ap = A + row * lda + koff;
    for (int k = 0; k < K; k += 4) {
        v2f a = *(const v2f*)(ap + k);                     // A[row][k+koff], A[row][k+koff+1]
        const float* bp = B + (k + koff) * ldb + col;
        v2f b;
        b.x = bp[0];                                       // B[k+koff][col]
        b.y = bp[ldb];                                     // B[k+koff+1][col]
        acc = __builtin_amdgcn_wmma_f32_16x16x4_f32(
            /*neg_a=*/false, a, /*neg_b=*/false, b,
            /*c_mod=*/(short)0, acc, /*reuse_a=*/false, /*reuse_b=*/false);
    }

    float bv = bias[col];
    int mbase = blockIdx.x * 16 + (half << 3);
#pragma unroll
    for (int r = 0; r < 8; r++) {
        float v = acc[r] + bv;
        if (relu) v = fmaxf(v, 0.f);
        C[(mbase + r) * ldc + col] = v;
    }
}

// ---------------- readout: mean of h over each graph (graph_ids sorted) ----------------
__global__ __launch_bounds__(256) void k_readout(const float* __restrict__ h,
                                                 const int* __restrict__ gid,
                                                 float* __restrict__ hg) {
    __shared__ float part[4][64];
    __shared__ float pcnt[4];
    int f = threadIdx.x & 63;
    int sub = threadIdx.x >> 6;
    int g = blockIdx.x;
    float acc = 0.f, cnt = 0.f;
    for (int n = sub; n < N_NODES; n += 4) {
        if (gid[n] == g) { acc += h[(n << 6) + f]; cnt += 1.f; }
    }
    part[sub][f] = acc;
    if (f == 0) pcnt[sub] = cnt;
    __syncthreads();
    if (sub == 0) {
        float s = part[0][f] + part[1][f] + part[2][f] + part[3][f];
        float c = pcnt[0] + pcnt[1] + pcnt[2] + pcnt[3];
        hg[(g << 6) + f] = s / fmaxf(c, 1.f);
    }
}

// ---------------- row L2 normalize of [G, EMB] ----------------
__global__ __launch_bounds__(128) void k_l2norm(float* __restrict__ out) {
    __shared__ float red[128];
    int g = blockIdx.x, t = threadIdx.x;
    float v = out[g * EMB + t];
    red[t] = v * v;
    __syncthreads();
    for (int s = 64; s > 0; s >>= 1) {
        if (t < s) red[t] += red[t + s];
        __syncthreads();
    }
    float nrm = fmaxf(sqrtf(red[0]), 1e-12f);
    out[g * EMB + t] = v / nrm;
}

// ---------------- launcher ----------------

extern "C" void kernel_launch(void* const* d_in, const int* in_sizes, int n_in,
                              void* d_out, int out_size, void* d_ws, size_t ws_size,
                              hipStream_t stream) {
    const float* h    = (const float*)d_in[0];
    const int*   src  = (const int*)d_in[1];
    const int*   dst  = (const int*)d_in[2];
    const int*   gid  = (const int*)d_in[3];
    const float* W[3]  = {(const float*)d_in[4], (const float*)d_in[6], (const float*)d_in[8]};
    const float* bb[3] = {(const float*)d_in[5], (const float*)d_in[7], (const float*)d_in[9]};
    const float* embW = (const float*)d_in[10];
    const float* embb = (const float*)d_in[11];
    float* out = (float*)d_out;

    float* feats = (float*)d_ws;                          // N*256
    float* xs    = feats + (size_t)N_NODES * 256;         // N*64
    float* bufA  = xs    + (size_t)N_NODES * 64;          // N*64
    float* bufB  = bufA  + (size_t)N_NODES * 64;          // N*64
    float* dn    = bufB  + (size_t)N_NODES * 64;          // N
    float* hg    = dn    + N_NODES;                       // 64*64

    dim3 b256(256);

    // degree -> deg_norm (in place in dn)
    k_zero<<<(N_NODES + 255) / 256, b256, 0, stream>>>(dn, N_NODES);
    k_degree<<<(N_EDGES + 255) / 256, b256, 0, stream>>>(dst, dn);
    k_degnorm<<<(N_NODES + 255) / 256, b256, 0, stream>>>(dn);

    const float* x = h;
    float* outs[3] = {bufA, bufB, bufA};
    for (int layer = 0; layer < 3; layer++) {
        k_prep<<<N_NODES, b256, 0, stream>>>(x, dn, feats, xs);   // N*256/256 == N_NODES blocks
        for (int hop = 1; hop <= HOPS; hop++) {
            int hopoff = hop * 64;
            k_scatter<<<(N_EDGES * 16 + 255) / 256, b256, 0, stream>>>(src, dst, xs, feats, hopoff);
            k_posthop<<<(N_NODES * 64 + 255) / 256, b256, 0, stream>>>(feats, dn, xs, hopoff);
        }
        dim3 grid(N_NODES / 16, HID / 16);
        k_wmma_gemm<<<grid, dim3(32), 0, stream>>>(feats, 256, W[layer], HID, bb[layer],
                                                   outs[layer], HID, 256, 1);
        x = outs[layer];
    }

    k_readout<<<N_GRAPHS, b256, 0, stream>>>(bufA, gid, hg);
    k_wmma_gemm<<<dim3(N_GRAPHS / 16, EMB / 16), dim3(32), 0, stream>>>(hg, HID, embW, EMB, embb,
                                                                        out, EMB, HID, 0);
    k_l2norm<<<N_GRAPHS, dim3(128), 0, stream>>>(out);
}